// MultiBoxLoss_90426241450786
// MI455X (gfx1250) — compile-verified
//
#include <hip/hip_runtime.h>
#include <math.h>

#define B_N 64
#define A_N 24564
#define C_N 81
#define O_N 32
#define TILE_A 2048
#define TB 128               // anchors per conf block
#define BPB 192              // ceil(A_N / TB)

#ifndef __has_builtin
#define __has_builtin(x) 0
#endif

#if __has_builtin(__builtin_amdgcn_global_load_async_to_lds_b128)
#define HAVE_ASYNC 1
#pragma message("CDNA5 probe: global_load_async_to_lds_b128 AVAILABLE")
#else
#define HAVE_ASYNC 0
#pragma message("CDNA5 probe: global_load_async_to_lds_b128 MISSING")
#endif

#if __has_builtin(__builtin_amdgcn_s_wait_asynccnt)
#define HAVE_WAITA 1
#else
#define HAVE_WAITA 0
#endif

#if __has_builtin(__builtin_amdgcn_wmma_f32_16x16x4_f32)
#define HAVE_WMMA4 1
#else
#define HAVE_WMMA4 0
#endif

typedef __attribute__((ext_vector_type(2))) float v2f;
typedef __attribute__((ext_vector_type(8))) float v8f;
typedef int v4i_t __attribute__((vector_size(16)));   // matches builtin param pointee

// ---------------- async copy helpers ----------------
__device__ __forceinline__ void async_cp16(float* l, const float* g) {
#if HAVE_ASYNC
    __builtin_amdgcn_global_load_async_to_lds_b128(
        (__attribute__((address_space(1))) v4i_t*)g,
        (__attribute__((address_space(3))) v4i_t*)l,
        0, 0);
#else
    *(float4*)l = *(const float4*)g;
#endif
}

__device__ __forceinline__ void wait_async0() {
#if HAVE_ASYNC
#if HAVE_WAITA
    __builtin_amdgcn_s_wait_asynccnt(0);
#else
    asm volatile("s_wait_asynccnt 0" ::: "memory");
#endif
#endif
}

// ---------------- block reductions ----------------
__device__ __forceinline__ float bredf(float v, float* buf) {
    int t = threadIdx.x;
    buf[t] = v; __syncthreads();
    for (int s = blockDim.x >> 1; s > 0; s >>= 1) {
        if (t < s) buf[t] += buf[t + s];
        __syncthreads();
    }
    float r = buf[0]; __syncthreads();
    return r;
}

__device__ __forceinline__ int bredi(int v, int* buf) {
    int t = threadIdx.x;
    buf[t] = v; __syncthreads();
    for (int s = blockDim.x >> 1; s > 0; s >>= 1) {
        if (t < s) buf[t] += buf[t + s];
        __syncthreads();
    }
    int r = buf[0]; __syncthreads();
    return r;
}

__device__ __forceinline__ unsigned long long shfl_xor_u64(unsigned long long v, int m) {
    unsigned lo = (unsigned)v, hi = (unsigned)(v >> 32);
    lo = __shfl_xor(lo, m, 32); hi = __shfl_xor(hi, m, 32);
    return ((unsigned long long)hi << 32) | lo;
}

// ============ Kernel A: matching + localization L1 (one block per batch) ============
__global__ __launch_bounds__(1024) void mb_match_kernel(
    const float* __restrict__ pred_locs,   // [B,A,4]
    const float* __restrict__ boxes,       // [B,O,4] xyxy
    const float* __restrict__ anchors,     // [A,4] cxcywh
    const int*   __restrict__ labels,      // [B,O]
    float* __restrict__ ovl_fa,            // [B,A]
    int*   __restrict__ obj_fa,            // [B,A]
    int*   __restrict__ true_cls,          // [B,A]
    int*   __restrict__ n_pos,             // [B]
    float* __restrict__ l1_sum)            // [B]
{
    __shared__ __align__(16) float s_anc[TILE_A * 4];
    __shared__ float s_box[O_N * 4];
    __shared__ int   s_lbl[O_N];
    __shared__ int   s_afo[O_N];
    __shared__ float s_redf[1024];
    __shared__ int   s_redi[1024];

    const int b = blockIdx.x;
    const int tid = threadIdx.x;
    const int lane = tid & 31;
    const int wave = tid >> 5;   // 32 waves == 32 objects

    if (tid < O_N * 4) s_box[tid] = boxes[(size_t)b * O_N * 4 + tid];
    if (tid < O_N)     s_lbl[tid] = labels[b * O_N + tid];
    __syncthreads();

    float* my_ovl = ovl_fa + (size_t)b * A_N;
    int*   my_obj = obj_fa + (size_t)b * A_N;

    // this wave's object box (for per-object argmax over anchors)
    const float bx1 = s_box[wave * 4 + 0], by1 = s_box[wave * 4 + 1];
    const float bx2 = s_box[wave * 4 + 2], by2 = s_box[wave * 4 + 3];
    const float barea = (bx2 - bx1) * (by2 - by1);
    unsigned long long bestkey = 0ull;

    for (int t0 = 0; t0 < A_N; t0 += TILE_A) {
        const int n = min(TILE_A, A_N - t0);
        // stage anchor tile into LDS (async path on CDNA5)
        for (int i = tid; i < n; i += 1024)
            async_cp16(&s_anc[i * 4], &anchors[(size_t)(t0 + i) * 4]);
        wait_async0();
        __syncthreads();

        // pass 1: per-anchor best object (first-index tie-break via strict >)
        for (int a = tid; a < n; a += 1024) {
            const float cx = s_anc[a*4], cy = s_anc[a*4+1], w = s_anc[a*4+2], h = s_anc[a*4+3];
            const float ax1 = cx - 0.5f*w, ay1 = cy - 0.5f*h, ax2 = cx + 0.5f*w, ay2 = cy + 0.5f*h;
            const float aarea = w * h;
            float best = -1.0f; int bo = 0;
            #pragma unroll 8
            for (int o = 0; o < O_N; ++o) {
                const float x1 = s_box[o*4], y1 = s_box[o*4+1], x2 = s_box[o*4+2], y2 = s_box[o*4+3];
                float iw = fminf(ax2, x2) - fmaxf(ax1, x1); iw = fmaxf(iw, 0.0f);
                float ih = fminf(ay2, y2) - fmaxf(ay1, y1); ih = fmaxf(ih, 0.0f);
                const float inter = iw * ih;
                const float oarea = (x2 - x1) * (y2 - y1);
                const float iou = inter / (oarea + aarea - inter);
                if (iou > best) { best = iou; bo = o; }
            }
            my_ovl[t0 + a] = best;
            my_obj[t0 + a] = bo;
        }

        // pass 2: this wave's object scans the tile for its best anchor
        for (int a = lane; a < n; a += 32) {
            const float cx = s_anc[a*4], cy = s_anc[a*4+1], w = s_anc[a*4+2], h = s_anc[a*4+3];
            const float ax1 = cx - 0.5f*w, ay1 = cy - 0.5f*h, ax2 = cx + 0.5f*w, ay2 = cy + 0.5f*h;
            float iw = fminf(ax2, bx2) - fmaxf(ax1, bx1); iw = fmaxf(iw, 0.0f);
            float ih = fminf(ay2, by2) - fmaxf(ay1, by1); ih = fmaxf(ih, 0.0f);
            const float inter = iw * ih;
            const float iou = inter / (barea + w * h - inter);
            const unsigned long long key =
                ((unsigned long long)__float_as_uint(iou) << 32) |
                (unsigned long long)(0xFFFFFFFFu - (unsigned)(t0 + a));
            if (key > bestkey) bestkey = key;
        }
        __syncthreads();
    }

    // wave-reduce argmax key; lane0 publishes anchor_fo[object]
    for (int s = 16; s > 0; s >>= 1) {
        unsigned long long o = shfl_xor_u64(bestkey, s);
        if (o > bestkey) bestkey = o;
    }
    if (lane == 0)
        s_afo[wave] = (int)(0xFFFFFFFFu - (unsigned)(bestkey & 0xFFFFFFFFull));
    __syncthreads();

    // serial override (matches JAX scatter last-wins on duplicates)
    if (tid == 0) {
        for (int o = 0; o < O_N; ++o) {
            const int af = s_afo[o];
            my_obj[af] = o;
            my_ovl[af] = 1.0f;
        }
    }
    __syncthreads();

    // pass 4: labels, positive count, localization L1
    int np = 0; float l1 = 0.0f;
    for (int a = tid; a < A_N; a += 1024) {
        const float ov = my_ovl[a];
        const int   o  = my_obj[a];
        const int lbl = (ov < 0.5f) ? 0 : s_lbl[o];
        true_cls[(size_t)b * A_N + a] = lbl;
        if (lbl != 0) {
            ++np;
            const float4 anc = ((const float4*)anchors)[a];
            const float x1 = s_box[o*4], y1 = s_box[o*4+1], x2 = s_box[o*4+2], y2 = s_box[o*4+3];
            const float bcx = 0.5f*(x1 + x2), bcy = 0.5f*(y1 + y2);
            const float bw = x2 - x1, bh = y2 - y1;
            const float g0 = (bcx - anc.x) * 10.0f / anc.z;
            const float g1 = (bcy - anc.y) * 10.0f / anc.w;
            const float g2 = logf(bw / anc.z) * 5.0f;
            const float g3 = logf(bh / anc.w) * 5.0f;
            const float4 pl = ((const float4*)pred_locs)[(size_t)b * A_N + a];
            l1 += fabsf(pl.x - g0) + fabsf(pl.y - g1) + fabsf(pl.z - g2) + fabsf(pl.w - g3);
        }
    }
    const float l1t = bredf(l1, s_redf);
    const int   npt = bredi(np, s_redi);
    if (tid == 0) { n_pos[b] = npt; l1_sum[b] = l1t; }
}

// ============ Kernel B: per-anchor CE via log-softmax (async LDS staged) ============
__global__ __launch_bounds__(TB) void mb_conf_kernel(
    const float* __restrict__ scores,     // [B,A,C]
    const int*   __restrict__ true_cls,   // [B,A]
    float* __restrict__ conf_neg,         // [B,A]  (0 for positives)
    float* __restrict__ pos_partials)     // [B*BPB]
{
    __shared__ __align__(16) float s_sc[TB * C_N];  // 41.5 KB
    __shared__ float s_red[TB];

    const int tid = threadIdx.x;
    const int b  = blockIdx.x / BPB;
    const int tb = blockIdx.x % BPB;
    const int a0 = tb * TB;
    const int na = min(TB, A_N - a0);
    const int count = na * C_N;            // always a multiple of 4
    const float* gbase = scores + (size_t)(b * A_N + a0) * C_N;  // 16B aligned

    // coalesced async copy of the whole tile into LDS
    const int nv = count >> 2;
    for (int v = tid; v < nv; v += TB)
        async_cp16(&s_sc[v * 4], gbase + (size_t)v * 4);
    wait_async0();
    __syncthreads();

    float pp = 0.0f;
    if (tid < na) {
        const int a = a0 + tid;
        const int cls = true_cls[(size_t)b * A_N + a];
        const float* row = &s_sc[tid * C_N];   // stride 81: LDS bank-conflict free
        float m = -3.402823466e+38f;
        #pragma unroll 3
        for (int j = 0; j < C_N; ++j) m = fmaxf(m, row[j]);
        float s = 0.0f;
        #pragma unroll 3
        for (int j = 0; j < C_N; ++j) s += expf(row[j] - m);
        const float conf = m + logf(s) - row[cls];
        const bool pos = (cls != 0);
        conf_neg[(size_t)b * A_N + a] = pos ? 0.0f : conf;
        pp = pos ? conf : 0.0f;
    }
    const float tile_pos = bredf(pp, s_red);
    if (tid == 0) pos_partials[blockIdx.x] = tile_pos;
}

// ============ Kernel C: per-batch top-k negative sum (binary search on bits) ============
__global__ __launch_bounds__(256) void mb_hardneg_kernel(
    const float* __restrict__ conf_neg,
    const float* __restrict__ pos_partials,
    const int*   __restrict__ n_pos,
    float* __restrict__ hard_out,
    float* __restrict__ pos_out)
{
    __shared__ float s_f[256];
    __shared__ int   s_i[256];
    __shared__ unsigned s_lo, s_hi;

    const int b = blockIdx.x;
    const int tid = threadIdx.x;

    // deterministic fixed-order sum of positive partials
    float ps = 0.0f;
    for (int i = tid; i < BPB; i += 256) ps += pos_partials[b * BPB + i];
    ps = bredf(ps, s_f);

    int k = 3 * n_pos[b];
    if (k > A_N) k = A_N;
    const unsigned* cn = (const unsigned*)conf_neg + (size_t)b * A_N;

    float S = 0.0f;
    if (k > 0) {
        if (tid == 0) { s_lo = 0u; s_hi = 0x7F800000u; }
        __syncthreads();
        for (;;) {
            const unsigned lo = s_lo, hi = s_hi;
            if (lo >= hi) break;
            const unsigned mid = lo + ((hi - lo + 1u) >> 1);
            int c = 0;
            for (int a = tid; a < A_N; a += 256) c += (cn[a] >= mid) ? 1 : 0;
            c = bredi(c, s_i);
            if (tid == 0) { if (c >= k) s_lo = mid; else s_hi = mid - 1u; }
            __syncthreads();
        }
        const unsigned tb = s_lo;
        const float tval = __uint_as_float(tb);
        int cg = 0; float sg = 0.0f;
        for (int a = tid; a < A_N; a += 256) {
            const unsigned v = cn[a];
            if (v > tb) { ++cg; sg += __uint_as_float(v); }
        }
        cg = bredi(cg, s_i);
        sg = bredf(sg, s_f);
        S = sg + (float)(k - cg) * tval;
    }
    if (tid == 0) { hard_out[b] = S; pos_out[b] = ps; }
}

// ============ Kernel D: final reduction over B=64 via WMMA f32 16x16x4 ============
__device__ __forceinline__ float sum64_wmma(float v0, float v1) {
#if HAVE_WMMA4
    v2f a;    a.x = v0;   a.y = v1;
    v2f ones; ones.x = 1.0f; ones.y = 1.0f;
    v8f c = {};
    v8f d = __builtin_amdgcn_wmma_f32_16x16x4_f32(false, a, false, ones,
                                                  (short)0, c, false, false);
    float s = d[0] + d[1] + d[2] + d[3] + d[4] + d[5] + d[6] + d[7];
    return s + __shfl_xor(s, 16, 32);
#else
    float s = v0 + v1;
    for (int m = 16; m > 0; m >>= 1) s += __shfl_xor(s, m, 32);
    return s;
#endif
}

__global__ __launch_bounds__(32) void mb_finalize_kernel(
    const float* __restrict__ hard,
    const float* __restrict__ pos,
    const float* __restrict__ l1s,
    const int*   __restrict__ n_pos,
    float* __restrict__ out)
{
    const int lane = threadIdx.x;
    // A-matrix (16x4 f32) layout: lanes 0-15 hold K=0,1; lanes 16-31 hold K=2,3
    const int m  = lane & 15;
    const int k0 = (lane < 16) ? 0 : 2;
    const int i0 = m * 4 + k0, i1 = i0 + 1;

    const float c0 = hard[i0] + pos[i0], c1 = hard[i1] + pos[i1];
    const float e0 = l1s[i0],            e1 = l1s[i1];
    const float p0 = (float)n_pos[i0],   p1 = (float)n_pos[i1];

    const float tot_conf = sum64_wmma(c0, c1);
    const float tot_l1   = sum64_wmma(e0, e1);
    const float tot_np   = sum64_wmma(p0, p1);

    if (lane == 0)
        out[0] = tot_conf / tot_np + tot_l1 / (4.0f * tot_np);
}

// ============================ host launcher ============================
extern "C" void kernel_launch(void* const* d_in, const int* in_sizes, int n_in,
                              void* d_out, int out_size, void* d_ws, size_t ws_size,
                              hipStream_t stream) {
    const float* pred_locs = (const float*)d_in[0];   // [B,A,4]
    const float* scores    = (const float*)d_in[1];   // [B,A,C]
    const float* boxes     = (const float*)d_in[2];   // [B,O,4]
    const float* anchors   = (const float*)d_in[3];   // [A,4]
    const int*   labels    = (const int*)d_in[4];     // [B,O]
    float* out = (float*)d_out;

    char* w = (char*)d_ws;
    const size_t BAb = (size_t)B_N * A_N * 4;         // 6,288,384 bytes (256-aligned)
    float* ws_ovl  = (float*)(w + 0 * BAb);
    int*   ws_obj  = (int*)  (w + 1 * BAb);
    int*   ws_cls  = (int*)  (w + 2 * BAb);
    float* ws_cneg = (float*)(w + 3 * BAb);
    char*  tail    = w + 4 * BAb;
    float* ws_part = (float*)(tail);                  // B_N*BPB = 12288 floats
    int*   ws_np   = (int*)  (tail + 49152);
    float* ws_l1   = (float*)(tail + 49152 + 256);
    float* ws_hard = (float*)(tail + 49152 + 512);
    float* ws_poss = (float*)(tail + 49152 + 768);

    mb_match_kernel<<<dim3(B_N), dim3(1024), 0, stream>>>(
        pred_locs, boxes, anchors, labels,
        ws_ovl, ws_obj, ws_cls, ws_np, ws_l1);

    mb_conf_kernel<<<dim3(B_N * BPB), dim3(TB), 0, stream>>>(
        scores, ws_cls, ws_cneg, ws_part);

    mb_hardneg_kernel<<<dim3(B_N), dim3(256), 0, stream>>>(
        ws_cneg, ws_part, ws_np, ws_hard, ws_poss);

    mb_finalize_kernel<<<dim3(1), dim3(32), 0, stream>>>(
        ws_hard, ws_poss, ws_l1, ws_np, out);
}